// NonSymNonLinearFieldChargesBlock_13872744366318
// MI455X (gfx1250) — compile-verified
//
#include <hip/hip_runtime.h>
#include <hip/hip_bf16.h>

typedef float v2f __attribute__((ext_vector_type(2)));
typedef float v8f __attribute__((ext_vector_type(8)));

#define CC 96
#define NN 4096

// workspace layout (floats); all weight panels stored TRANSPOSED: WT[c*K + k]
#define OFF_A    0                       // A^T     (96x96)
#define OFF_B    (OFF_A + CC * CC)       // B^T
#define OFF_C    (OFF_B + CC * CC)       // Cmat^T
#define OFF_L10  (OFF_C + CC * CC)       // Wl10^T
#define OFF_L11  (OFF_L10 + CC * CC)     // Wl11^T
#define OFF_O0A  (OFF_L11 + CC * CC)     // Wout0a^T
#define OFF_O0B  (OFF_O0A + CC * CC)     // Wout0b^T
#define OFF_O1A  (OFF_O0B + CC * CC)     // Wout1a^T
#define OFF_O1B  (OFF_O1A + CC * CC)     // Wout1b^T
#define OFF_M1   (OFF_O1B + CC * CC)     // Wmlp1^T (16x96)
#define OFF_M2   (OFF_M1 + 16 * CC)      // Wmlp2^T (16x16)
#define OFF_M3   (OFF_M2 + 16 * 16)      // Wmlp3^T (16x16)
#define OFF_M4   (OFF_M3 + 16 * 16)      // Wmlp4^T (384x16)
#define WS_TOTAL (OFF_M4 + 4 * CC * 16)  // 91136 floats = 356 KB

__device__ __constant__ float INV_SQRT_C  = 0.1020620726f;   // 1/sqrt(96)
__device__ __constant__ float INV_SQRT3   = 0.5773502692f;   // 1/sqrt(3)
__device__ __constant__ float INV_SQRT2CC = 0.0073656935f;   // 1/(96*sqrt(2))
__device__ __constant__ float NORM_OUT    = 0.0721687836f;   // 1/sqrt(192)
__device__ __constant__ float SILU_CST    = 1.6761612f;      // 1/sqrt(E[silu(z)^2])

static __device__ __forceinline__ v8f wmma4(v2f a, v2f b, v8f c) {
  // D(16x16,f32) = A(16x4,f32) * B(4x16,f32) + C  -> v_wmma_f32_16x16x4_f32
  return __builtin_amdgcn_wmma_f32_16x16x4_f32(false, a, false, b, (short)0, c,
                                               false, false);
}

static __device__ __forceinline__ float silu_n(float x) {
  return SILU_CST * (x / (1.0f + __expf(-x)));
}

// B-fragment from transposed panel (K = contraction length): one b64 load.
static __device__ __forceinline__ v2f bfragT(const float* __restrict__ WT,
                                             int c, int kk, int K) {
  return *(const v2f*)(WT + (size_t)c * K + kk);
}

// ---------------------------------------------------------------------------
// Setup 1: collapse rank-1 channel structure of y0/y1 into the C^3 tensors,
// writing results transposed:  AT[w*C+u] = sum_v Wl20[v] * W00[u,v,w]  etc.
// ---------------------------------------------------------------------------
__global__ void precompute_kernel(const float* __restrict__ Wf00,
                                  const float* __restrict__ Wf11,
                                  const float* __restrict__ Wl20,
                                  const float* __restrict__ bl20,
                                  const float* __restrict__ Wl21,
                                  float* __restrict__ ws) {
  int t = blockIdx.x * blockDim.x + threadIdx.x;
  if (t < CC * CC) {
    int u = t / CC, w = t % CC;
    const float* base = Wf00 + (size_t)u * CC * CC + w;
    float a = 0.f, b = 0.f;
    for (int v = 0; v < CC; ++v) {
      float x = base[v * CC];
      a += Wl20[v] * x;
      b += bl20[v] * x;
    }
    ws[OFF_A + w * CC + u] = a;
    ws[OFF_B + w * CC + u] = b;
  } else if (t < 2 * CC * CC) {
    int t2 = t - CC * CC;
    int u = t2 / CC, w = t2 % CC;
    const float* base = Wf11 + (size_t)u * CC * CC + w;
    float c = 0.f;
    for (int v = 0; v < CC; ++v) c += Wl21[v] * base[v * CC];
    ws[OFF_C + w * CC + u] = c;
  }
}

// Setup 2: generic transpose  dst[c*R + r] = src[r*Ncol + c]
__global__ void transpose_kernel(const float* __restrict__ src,
                                 float* __restrict__ dst, int R, int Ncol) {
  int t = blockIdx.x * blockDim.x + threadIdx.x;
  if (t < R * Ncol) {
    int r = t / Ncol, c = t % Ncol;
    dst[c * R + r] = src[t];
  }
}

// ---------------------------------------------------------------------------
// Fused per-node pipeline: one wave32 per 16-node tile, LDS-resident
// activations (< 64 KB), all matrix math via v_wmma_f32_16x16x4_f32.
// ---------------------------------------------------------------------------
__global__ void __launch_bounds__(32) fused_kernel(
    const float* __restrict__ nf, const float* __restrict__ pf,
    const float* __restrict__ ws, float* __restrict__ out) {
  const int lane = threadIdx.x;       // wave32
  const int row  = lane & 15;         // A-frag row / B-frag col / D col
  const int kh   = lane >> 4;         // K half-select
  const int n0   = blockIdx.x * 16;   // 16 nodes per wave

  const float* Wl10T = ws + OFF_L10;
  const float* Wl11T = ws + OFF_L11;
  const float* AT    = ws + OFF_A;
  const float* BT    = ws + OFF_B;
  const float* CT    = ws + OFF_C;
  const float* Wm1T  = ws + OFF_M1;
  const float* Wm2T  = ws + OFF_M2;
  const float* Wm3T  = ws + OFF_M3;
  const float* Wm4T  = ws + OFF_M4;
  const float* Wo0aT = ws + OFF_O0A;
  const float* Wo0bT = ws + OFF_O0B;
  const float* Wo1aT = ws + OFF_O1A;
  const float* Wo1bT = ws + OFF_O1B;

  __shared__ __align__(16) float sX0[16 * CC];      // x0 after l1        (6 KB)
  __shared__ __align__(16) float sX1[3 * 16 * CC];  // x1 after l1        (18 KB)
  __shared__ __align__(16) float sQ [16 * CC];      // q = sum_m x1*p1    (6 KB)
  __shared__ __align__(16) float sNm[16 * CC];      // norms              (6 KB)
  __shared__ __align__(16) float sTP[16 * 4 * CC];  // nf stage, then tp_w(24 KB)
  __shared__ __align__(16) float sHa[16 * 16];      // MLP ping           (1 KB)
  __shared__ __align__(16) float sHb[16 * 16];      // MLP pong           (1 KB)
  __shared__ __align__(16) float sP [16 * 4];       // pot feats

  // ---- stage 0: stage the 16x384 node-feature tile (coalesced b128) + pot ----
  for (int i = lane * 4; i < 16 * 4 * CC; i += 32 * 4)
    *(float4*)(sTP + i) = *(const float4*)(nf + (size_t)n0 * 4 * CC + i);
  {
    int i = lane * 2;
    sP[i]     = pf[n0 * 4 + i];
    sP[i + 1] = pf[n0 * 4 + i + 1];
  }
  __syncthreads();

  // ---- stage 1: x0 = (nf[:, :C] @ Wl10) * inv_sqrtC ----
  for (int c0 = 0; c0 < CC; c0 += 16) {
    v8f acc = {0.f, 0.f, 0.f, 0.f, 0.f, 0.f, 0.f, 0.f};
    for (int k = 0; k < CC; k += 4) {
      int kk = k + 2 * kh;
      v2f a = *(const v2f*)(sTP + row * 4 * CC + kk);
      acc = wmma4(a, bfragT(Wl10T, c0 + row, kk, CC), acc);
    }
    for (int r = 0; r < 8; ++r)
      sX0[(r + 8 * kh) * CC + c0 + row] = acc[r] * INV_SQRT_C;
  }

  // ---- stage 2: x1[:,:,m] = (nf[:,C:].reshape(.,C,3)[:,:,m] @ Wl11) * inv_sqrtC ----
  for (int m = 0; m < 3; ++m)
    for (int c0 = 0; c0 < CC; c0 += 16) {
      v8f acc = {0.f, 0.f, 0.f, 0.f, 0.f, 0.f, 0.f, 0.f};
      const float* nr = sTP + row * 4 * CC + CC;  // staged x1 input (u,m interleaved)
      for (int k = 0; k < CC; k += 4) {
        int kk = k + 2 * kh;
        v2f a; a.x = nr[3 * kk + m]; a.y = nr[3 * (kk + 1) + m];
        acc = wmma4(a, bfragT(Wl11T, c0 + row, kk, CC), acc);
      }
      for (int r = 0; r < 8; ++r)
        sX1[m * 16 * CC + (r + 8 * kh) * CC + c0 + row] = acc[r] * INV_SQRT_C;
    }
  __syncthreads();

  // ---- stage 3: q[n,u] = sum_m x1[n,u,m]*p1[n,m] (reused by z1 AND m0b) ----
  for (int i = lane; i < 16 * CC; i += 32) {
    int r = i / CC;
    sQ[i] = sX1[i]           * sP[r * 4 + 1]
          + sX1[16 * CC + i] * sP[r * 4 + 2]
          + sX1[32 * CC + i] * sP[r * 4 + 3];
  }
  __syncthreads();

  // ---- stage 4: z0 = p0*(x0@A)+x0@B ; z1 = (q@Cm)/sqrt3 ; norms = product^2 ----
  for (int c0 = 0; c0 < CC; c0 += 16) {
    v8f aA = {0.f,0.f,0.f,0.f,0.f,0.f,0.f,0.f};
    v8f aB = {0.f,0.f,0.f,0.f,0.f,0.f,0.f,0.f};
    v8f aC = {0.f,0.f,0.f,0.f,0.f,0.f,0.f,0.f};
    for (int k = 0; k < CC; k += 4) {
      int kk = k + 2 * kh;
      v2f a0 = *(const v2f*)(sX0 + row * CC + kk);
      v2f aq = *(const v2f*)(sQ  + row * CC + kk);
      aA = wmma4(a0, bfragT(AT, c0 + row, kk, CC), aA);
      aB = wmma4(a0, bfragT(BT, c0 + row, kk, CC), aB);
      aC = wmma4(aq, bfragT(CT, c0 + row, kk, CC), aC);
    }
    for (int r = 0; r < 8; ++r) {
      int mr = r + 8 * kh;
      float prod = (sP[mr * 4] * aA[r] + aB[r] + aC[r] * INV_SQRT3) * INV_SQRT2CC;
      sNm[mr * CC + c0 + row] = prod * prod;
    }
  }
  __syncthreads();

  // ---- stage 5: 4-layer MLP (96->16->16->16->384) ----
  {  // h1 = CST*silu(norms @ Wm1 * inv_sqrtC)
    v8f acc = {0.f,0.f,0.f,0.f,0.f,0.f,0.f,0.f};
    for (int k = 0; k < CC; k += 4) {
      int kk = k + 2 * kh;
      v2f a = *(const v2f*)(sNm + row * CC + kk);
      acc = wmma4(a, bfragT(Wm1T, row, kk, CC), acc);
    }
    for (int r = 0; r < 8; ++r)
      sHa[(r + 8 * kh) * 16 + row] = silu_n(acc[r] * INV_SQRT_C);
  }
  __syncthreads();
  {  // h2 = CST*silu(h1 @ Wm2 * 0.25)
    v8f acc = {0.f,0.f,0.f,0.f,0.f,0.f,0.f,0.f};
    for (int k = 0; k < 16; k += 4) {
      int kk = k + 2 * kh;
      v2f a = *(const v2f*)(sHa + row * 16 + kk);
      acc = wmma4(a, bfragT(Wm2T, row, kk, 16), acc);
    }
    for (int r = 0; r < 8; ++r)
      sHb[(r + 8 * kh) * 16 + row] = silu_n(acc[r] * 0.25f);
  }
  __syncthreads();
  {  // h3 = CST*silu(h2 @ Wm3 * 0.25)
    v8f acc = {0.f,0.f,0.f,0.f,0.f,0.f,0.f,0.f};
    for (int k = 0; k < 16; k += 4) {
      int kk = k + 2 * kh;
      v2f a = *(const v2f*)(sHb + row * 16 + kk);
      acc = wmma4(a, bfragT(Wm3T, row, kk, 16), acc);
    }
    for (int r = 0; r < 8; ++r)
      sHa[(r + 8 * kh) * 16 + row] = silu_n(acc[r] * 0.25f);
  }
  __syncthreads();
  for (int c0 = 0; c0 < 4 * CC; c0 += 16) {  // tp_w = h3 @ Wm4 * 0.25 (overwrites nf stage)
    v8f acc = {0.f,0.f,0.f,0.f,0.f,0.f,0.f,0.f};
    for (int k = 0; k < 16; k += 4) {
      int kk = k + 2 * kh;
      v2f a = *(const v2f*)(sHa + row * 16 + kk);
      acc = wmma4(a, bfragT(Wm4T, c0 + row, kk, 16), acc);
    }
    for (int r = 0; r < 8; ++r)
      sTP[(r + 8 * kh) * 4 * CC + c0 + row] = acc[r] * 0.25f;
  }
  __syncthreads();

  // ---- stage 6: out0 = (m0a@Wo0a + m0b@Wo0b)*norm_out, frags built on the fly ----
  for (int c0 = 0; c0 < CC; c0 += 16) {
    v8f acc = {0.f,0.f,0.f,0.f,0.f,0.f,0.f,0.f};
    float p0r = sP[row * 4];
    for (int k = 0; k < CC; k += 4) {
      int kk = k + 2 * kh;
      v2f w0 = *(const v2f*)(sTP + row * 4 * CC + kk);           // w00
      v2f w3 = *(const v2f*)(sTP + row * 4 * CC + 3 * CC + kk);  // w11
      v2f x0 = *(const v2f*)(sX0 + row * CC + kk);
      v2f qq = *(const v2f*)(sQ  + row * CC + kk);
      v2f a1; a1.x = w0.x * x0.x * p0r;       a1.y = w0.y * x0.y * p0r;
      v2f a2; a2.x = w3.x * qq.x * INV_SQRT3; a2.y = w3.y * qq.y * INV_SQRT3;
      acc = wmma4(a1, bfragT(Wo0aT, c0 + row, kk, CC), acc);
      acc = wmma4(a2, bfragT(Wo0bT, c0 + row, kk, CC), acc);
    }
    for (int r = 0; r < 8; ++r) {
      int mr = r + 8 * kh;
      out[(size_t)(n0 + mr) * 4 * CC + c0 + row] = acc[r] * NORM_OUT;
    }
  }

  // ---- stage 7: out1[:,v,m] = (m1a[:,:,m]@Wo1a + m1b[:,:,m]@Wo1b)*norm_out ----
  for (int m = 0; m < 3; ++m)
    for (int c0 = 0; c0 < CC; c0 += 16) {
      v8f acc = {0.f,0.f,0.f,0.f,0.f,0.f,0.f,0.f};
      float p1m = sP[row * 4 + 1 + m];
      float p0r = sP[row * 4];
      for (int k = 0; k < CC; k += 4) {
        int kk = k + 2 * kh;
        v2f w1 = *(const v2f*)(sTP + row * 4 * CC + CC + kk);      // w01
        v2f w2 = *(const v2f*)(sTP + row * 4 * CC + 2 * CC + kk);  // w10
        v2f x0 = *(const v2f*)(sX0 + row * CC + kk);
        v2f x1 = *(const v2f*)(sX1 + m * 16 * CC + row * CC + kk);
        v2f a1; a1.x = w1.x * x0.x * p1m; a1.y = w1.y * x0.y * p1m;
        v2f a2; a2.x = w2.x * x1.x * p0r; a2.y = w2.y * x1.y * p0r;
        acc = wmma4(a1, bfragT(Wo1aT, c0 + row, kk, CC), acc);
        acc = wmma4(a2, bfragT(Wo1bT, c0 + row, kk, CC), acc);
      }
      for (int r = 0; r < 8; ++r) {
        int mr = r + 8 * kh;
        // out layout: [out0 (C) | out1.reshape(3C): idx = C + 3*v + m]
        out[(size_t)(n0 + mr) * 4 * CC + CC + 3 * (c0 + row) + m] = acc[r] * NORM_OUT;
      }
    }
}

extern "C" void kernel_launch(void* const* d_in, const int* in_sizes, int n_in,
                              void* d_out, int out_size, void* d_ws, size_t ws_size,
                              hipStream_t stream) {
  const float* nf   = (const float*)d_in[0];
  const float* pf   = (const float*)d_in[1];
  const float* Wl10 = (const float*)d_in[2];
  const float* Wl11 = (const float*)d_in[3];
  const float* Wl20 = (const float*)d_in[4];
  const float* bl20 = (const float*)d_in[5];
  const float* Wl21 = (const float*)d_in[6];
  const float* Wf00 = (const float*)d_in[7];
  const float* Wf11 = (const float*)d_in[8];
  const float* Wm1  = (const float*)d_in[9];
  const float* Wm2  = (const float*)d_in[10];
  const float* Wm3  = (const float*)d_in[11];
  const float* Wm4  = (const float*)d_in[12];
  const float* Wo0a = (const float*)d_in[13];
  const float* Wo0b = (const float*)d_in[14];
  const float* Wo1a = (const float*)d_in[15];
  const float* Wo1b = (const float*)d_in[16];
  float* out = (float*)d_out;
  float* ws  = (float*)d_ws;

  // one-time setup: tensor collapse (A,B,Cmat) + transposed weight panels
  precompute_kernel<<<(2 * CC * CC + 255) / 256, 256, 0, stream>>>(
      Wf00, Wf11, Wl20, bl20, Wl21, ws);
  const int TB = 256;
  const int nCC = (CC * CC + TB - 1) / TB;
  transpose_kernel<<<nCC, TB, 0, stream>>>(Wl10, ws + OFF_L10, CC, CC);
  transpose_kernel<<<nCC, TB, 0, stream>>>(Wl11, ws + OFF_L11, CC, CC);
  transpose_kernel<<<nCC, TB, 0, stream>>>(Wo0a, ws + OFF_O0A, CC, CC);
  transpose_kernel<<<nCC, TB, 0, stream>>>(Wo0b, ws + OFF_O0B, CC, CC);
  transpose_kernel<<<nCC, TB, 0, stream>>>(Wo1a, ws + OFF_O1A, CC, CC);
  transpose_kernel<<<nCC, TB, 0, stream>>>(Wo1b, ws + OFF_O1B, CC, CC);
  transpose_kernel<<<(16 * CC + TB - 1) / TB, TB, 0, stream>>>(Wm1, ws + OFF_M1, CC, 16);
  transpose_kernel<<<1, TB, 0, stream>>>(Wm2, ws + OFF_M2, 16, 16);
  transpose_kernel<<<1, TB, 0, stream>>>(Wm3, ws + OFF_M3, 16, 16);
  transpose_kernel<<<(16 * 4 * CC + TB - 1) / TB, TB, 0, stream>>>(Wm4, ws + OFF_M4, 16, 4 * CC);

  fused_kernel<<<NN / 16, 32, 0, stream>>>(nf, pf, ws, out);
}